// MappingNetwork_20358144983686
// MI455X (gfx1250) — compile-verified
//
#include <hip/hip_runtime.h>
#include <stdint.h>

// Reference: buckets = linspace(-1e5, 1e5, 1e8) in f32;
// seeds = searchsorted(buckets, z[:,0], side='left'); out = broadcast(seeds, (B,512))
// Output dtype: int32, 16384*512 elements.

#define VMIN_F      (-100000.0f)
#define NUM_BUCKETS 100000000
#define REPEAT_W    512

// f32 step of the linspace: float(200000.0/99999999.0) == float(0.002)
__device__ __forceinline__ int searchsorted_uniform_f32(float zv) {
#pragma clang fp contract(off)
    const float step = 0.002f;
    int lo = 0, hi = NUM_BUCKETS;
    // Monotone predicate: bucket(m) = rn(rn(float(m)*step) + VMIN) >= zv
    // 27 iterations cover 1e8 (2^27 > 1e8). contract(off) keeps the
    // two-rounding mul+add (no v_fmaak contraction) so the bucket function
    // matches the f32 linspace model exactly.
    #pragma unroll 1
    while (lo < hi) {
        int mid = lo + ((hi - lo) >> 1);
        float b = __fadd_rn(__fmul_rn((float)mid, step), VMIN_F);
        if (b < zv) lo = mid + 1;
        else        hi = mid;
    }
    return lo;  // in [0, NUM_BUCKETS]
}

// Phase 1: one thread per row; z[:,0] is a strided (2 KB) gather.
__global__ void mapping_seeds_kernel(const float* __restrict__ z,
                                     int* __restrict__ seeds, int rows) {
    int i = blockIdx.x * blockDim.x + threadIdx.x;
    if (i < rows) {
        float zv = z[(size_t)i * REPEAT_W];
        seeds[i] = searchsorted_uniform_f32(zv);
    }
}

// Phase 2: one block per row; 128 lanes * int4 = 512 int32 = 2 KB per row.
// Stage the row in LDS, then push it out with the CDNA5 async LDS->global
// store engine (ASYNCcnt path). Store-bandwidth bound either way; this keeps
// the write stream on the gfx1250 async path.
__global__ void mapping_bcast_kernel(const int* __restrict__ seeds,
                                     int* __restrict__ out) {
    const int row = blockIdx.x;
    const int tid = threadIdx.x;          // 0..127, 4 waves (wave32)
    const int s   = seeds[row];           // wave-uniform -> s_load_b32

    __shared__ int4 lds4[128];
    int4 v;
    v.x = s; v.y = s; v.z = s; v.w = s;
    lds4[tid] = v;                        // ds_store_b128
    __syncthreads();                      // waits DScnt, barrier

    int* gaddr = out + (size_t)row * REPEAT_W + (size_t)tid * 4;
    // Low 32 bits of the flat LDS pointer == workgroup-relative LDS byte offset,
    // which is what VSRC of the async store expects.
    unsigned loff = (unsigned)(uintptr_t)(&lds4[tid]);

    asm volatile("global_store_async_from_lds_b128 %0, %1, off"
                 :
                 : "v"(gaddr), "v"(loff)
                 : "memory");
    asm volatile("s_wait_asynccnt 0" ::: "memory");
}

extern "C" void kernel_launch(void* const* d_in, const int* in_sizes, int n_in,
                              void* d_out, int out_size, void* d_ws, size_t ws_size,
                              hipStream_t stream) {
    const float* z = (const float*)d_in[0];   // (rows, 512) f32
    // d_in[1] (c) unused by the forward pass.
    int* out   = (int*)d_out;                  // (rows, 512) int32
    int* seeds = (int*)d_ws;                   // rows * 4 bytes of scratch

    const int rows = in_sizes[0] / REPEAT_W;   // 16384

    // Phase 1: compute one bucket index per row.
    {
        const int threads = 256;
        const int blocks  = (rows + threads - 1) / threads;
        mapping_seeds_kernel<<<blocks, threads, 0, stream>>>(z, seeds, rows);
    }

    // Phase 2: broadcast each index across 512 columns (32 MiB streaming store).
    {
        mapping_bcast_kernel<<<rows, 128, 0, stream>>>(seeds, out);
    }
}